// GroupingPointNetLayer_54640573940067
// MI455X (gfx1250) — compile-verified
//
#include <hip/hip_runtime.h>

typedef __attribute__((ext_vector_type(2))) float v2f;
typedef __attribute__((ext_vector_type(8))) float v8f;

#define KB     4
#define RPTS   16384
#define KAPPA  32
#define NFEAT  16          // NX + NF = 3 + 13
#define DOUT   64
#define NPTS   (KB * RPTS) // 65536
#define PH_STRIDE 66       // padded LDS row stride (floats) to avoid bank conflicts

// ---------------------------------------------------------------------------
// Kernel 1: pack P = [X | F] into contiguous 16-float (64B) rows in workspace,
// so the neighbor gather in the main kernel is a clean b64 load per lane.
// ---------------------------------------------------------------------------
__global__ void gpn_pack(const float* __restrict__ X, const float* __restrict__ F,
                         float* __restrict__ P) {
    int t = blockIdx.x * blockDim.x + threadIdx.x;   // over NPTS*16
    if (t >= NPTS * NFEAT) return;
    int row = t >> 4, j = t & 15;
    P[t] = (j < 3) ? X[row * 3 + j] : F[row * 13 + (j - 3)];
}

// ---------------------------------------------------------------------------
// Kernel 2: gather -> WMMA GEMM1 (H^T) -> relu -> neighbor max -> WMMA GEMM2
// (gamma) -> +bias -> relu.  One wave per 16-point tile.
// ---------------------------------------------------------------------------
__global__ __launch_bounds__(256)
void gpn_main(const float* __restrict__ P, const int* __restrict__ N,
              const float* __restrict__ H, const float* __restrict__ G,
              const float* __restrict__ bias, float* __restrict__ out) {
    __shared__ float gm[DOUT * DOUT];          // gamma, shared by all 8 waves
    __shared__ float ph[8][16 * PH_STRIDE];    // per-wave PHT_max staging tile

    const int tid  = threadIdx.x;
    const int lane = tid & 31;
    const int wave = tid >> 5;
    const int l15  = lane & 15;
    const int hi   = (lane >> 4) & 1;          // 0: lanes 0-15, 1: lanes 16-31

    // Cooperative load of gamma into LDS.
    for (int i = tid; i < DOUT * DOUT; i += 256) gm[i] = G[i];
    __syncthreads();

    // Stage-1 B matrix = H^T, kept resident in VGPRs.
    // B[k=feature][n=dout]; frag(t,kk): lane holds column d=16t+l15,
    // V0=B[ks][d]=H[d][ks], V1=H[d][ks+1], ks = 4kk + 2*hi.
    v2f b1[4][4];
#pragma unroll
    for (int t = 0; t < 4; ++t)
#pragma unroll
        for (int kk = 0; kk < 4; ++kk) {
            int d  = 16 * t + l15;
            int ks = 4 * kk + 2 * hi;
            b1[t][kk] = *(const v2f*)(H + d * NFEAT + ks);
        }

    const int pb = blockIdx.x * 8 + wave;      // 16-point tile id (0..4095)
    const int g0 = pb * 16;                    // first global point index
    float* phw = &ph[wave][0];

    // ---------------- Stage 1: per-point neighbor GEMM + relu + max --------
    for (int p = 0; p < 16; ++p) {
        const int  g  = g0 + p;
        const int  kb = g / RPTS;
        const long brow = (long)kb * RPTS;
        const int* nrow = N + (long)g * KAPPA;

        float cmax[4] = {-3.4e38f, -3.4e38f, -3.4e38f, -3.4e38f};

#pragma unroll
        for (int mt = 0; mt < 2; ++mt) {       // neighbors 0-15, 16-31
            int idx = nrow[mt * 16 + l15];     // lane = A-matrix row
            const float* arow = P + (brow + idx) * NFEAT;
            v8f acc[4] = {};                   // 4 N-tiles of 16x16 f32
#pragma unroll
            for (int kk = 0; kk < 4; ++kk) {   // K = 16 via 4 x (k=4)
                v2f a = *(const v2f*)(arow + 4 * kk + 2 * hi);
#pragma unroll
                for (int t = 0; t < 4; ++t)
                    acc[t] = __builtin_amdgcn_wmma_f32_16x16x4_f32(
                        false, a, false, b1[t][kk], (short)0, acc[t], false, false);
            }
            // per-lane max over this tile's 8 rows
#pragma unroll
            for (int t = 0; t < 4; ++t) {
                float m = acc[t][0];
#pragma unroll
                for (int j = 1; j < 8; ++j) m = fmaxf(m, acc[t][j]);
                cmax[t] = fmaxf(cmax[t], m);
            }
        }
        // combine the other half-wave's 16 rows, relu, stash in LDS
#pragma unroll
        for (int t = 0; t < 4; ++t) {
            float o = __shfl_xor(cmax[t], 16, 32);
            float v = fmaxf(fmaxf(cmax[t], o), 0.0f);
            if (hi == 0) phw[p * PH_STRIDE + 16 * t + l15] = v;
        }
    }

    // ---------------- Stage 2: [16 x 64] @ gamma[64 x 64] + bias, relu -----
    v8f acc2[4] = {};
#pragma unroll 4
    for (int kk = 0; kk < 16; ++kk) {          // K = 64 via 16 x (k=4)
        int ks = 4 * kk + 2 * hi;
        v2f a = *(const v2f*)(phw + l15 * PH_STRIDE + ks);  // A row = point
#pragma unroll
        for (int t = 0; t < 4; ++t) {
            int e = 16 * t + l15;
            v2f b;
            b.x = gm[ks * DOUT + e];
            b.y = gm[(ks + 1) * DOUT + e];
            acc2[t] = __builtin_amdgcn_wmma_f32_16x16x4_f32(
                false, a, false, b, (short)0, acc2[t], false, false);
        }
    }

    // bias + relu + store (C layout: lane = column, VGPR j / half = row)
#pragma unroll
    for (int t = 0; t < 4; ++t) {
        int e = 16 * t + l15;
        float bb = bias[e];
#pragma unroll
        for (int j = 0; j < 8; ++j) {
            int m = j + 8 * hi;
            out[(long)(g0 + m) * DOUT + e] = fmaxf(acc2[t][j] + bb, 0.0f);
        }
    }
}

// ---------------------------------------------------------------------------
extern "C" void kernel_launch(void* const* d_in, const int* in_sizes, int n_in,
                              void* d_out, int out_size, void* d_ws, size_t ws_size,
                              hipStream_t stream) {
    const float* X    = (const float*)d_in[0];
    const float* F    = (const float*)d_in[1];
    const int*   Nidx = (const int*)d_in[2];
    const float* H    = (const float*)d_in[3];
    const float* G    = (const float*)d_in[4];
    const float* bias = (const float*)d_in[5];
    float* out = (float*)d_out;

    float* Ppack = (float*)d_ws;               // NPTS * 16 floats = 4 MB

    int packN = NPTS * NFEAT;
    gpn_pack<<<(packN + 255) / 256, 256, 0, stream>>>(X, F, Ppack);

    // 4096 point-tiles of 16 points, 8 waves per 256-thread block
    gpn_main<<<(NPTS / 16) / 8, 256, 0, stream>>>(Ppack, Nidx, H, G, bias, out);
}